// GraphCondGlobal_652835029230
// MI455X (gfx1250) — compile-verified
//
#include <hip/hip_runtime.h>

#define N_NODES 50000
#define N_EDGES 800000
#define D_H     128
#define D_COND  16
#define N_C     2
#define N_GRAPHS 64

typedef __attribute__((ext_vector_type(2))) float v2f;
typedef __attribute__((ext_vector_type(8))) float v8f;

// ---------------------------------------------------------------- fill
__global__ void fill_f32(float* __restrict__ p, float v, long n) {
    long i = (long)blockIdx.x * blockDim.x + threadIdx.x;
    if (i < n) p[i] = v;
}

// ---------------------------------------------------------------- degree / normalization
__global__ void deg_kernel(const int* __restrict__ dst, float* __restrict__ deg, int nE) {
    int e = blockIdx.x * blockDim.x + threadIdx.x;
    if (e < nE) atomicAdd(&deg[dst[e]], 1.0f);
}

__global__ void dis_kernel(float* __restrict__ deg_dis, int n) {
    int i = blockIdx.x * blockDim.x + threadIdx.x;
    if (i < n) deg_dis[i] = rsqrtf(deg_dis[i]);   // deg pre-initialized to 1.0 (self loop)
}

__global__ void norm_kernel(const int* __restrict__ src, const int* __restrict__ dst,
                            const float* __restrict__ dis, float* __restrict__ norm, int nE) {
    int e = blockIdx.x * blockDim.x + threadIdx.x;
    if (e < nE) norm[e] = dis[src[e]] * dis[dst[e]];
}

// ---------------------------------------------------------------- GEMM via V_WMMA_F32_16X16X4_F32
// out[M x 128] = act(H)[M x 128] @ W[128 x 128]   (M multiple of 16)
// ReLU of the previous layer is fused into the LDS staging load (uniform flag,
// no divergence -> EXEC all-ones at every WMMA as the ISA requires).
// One block = 256 threads = 8 waves; block handles 16 rows x all 128 cols;
// wave w computes the 16x16 tile at columns [16w, 16w+16).
__global__ __launch_bounds__(256) void gemm_wmma_f32(
    const float* __restrict__ H, const float* __restrict__ W,
    float* __restrict__ out, int relu_in) {
    __shared__ float lds[16 * 132];            // padded stride: banks 0,4,...,60 per row
    const int m0 = blockIdx.x << 4;
    const int t  = threadIdx.x;

    for (int i = t; i < 16 * 128; i += 256) {  // tile rows contiguous -> coalesced
        int r = i >> 7, c = i & 127;
        float v = H[(size_t)m0 * 128 + i];
        if (relu_in) v = fmaxf(v, 0.0f);
        lds[r * 132 + c] = v;
    }
    __syncthreads();

    const int lane = t & 31;
    const int wave = t >> 5;
    const int rn   = lane & 15;                // m (A) / n (B) within tile
    const int kg   = (lane >> 4) << 1;         // K-group: 0 or 2  (ISA 16x4 f32 layout)
    const int n0   = wave << 4;

    v8f c = {};
    #pragma unroll
    for (int k0 = 0; k0 < 128; k0 += 4) {
        v2f a, b;
        a.x = lds[rn * 132 + k0 + kg];                       // A[m][k0+kg]
        a.y = lds[rn * 132 + k0 + kg + 1];                   // A[m][k0+kg+1]
        b.x = W[(size_t)(k0 + kg) * 128 + n0 + rn];          // B[k0+kg][n]
        b.y = W[(size_t)(k0 + kg + 1) * 128 + n0 + rn];      // B[k0+kg+1][n]
        c = __builtin_amdgcn_wmma_f32_16x16x4_f32(
                false, a, false, b, (short)0, c, false, false);
    }

    const int half = lane >> 4;
    const int n    = n0 + rn;
    #pragma unroll
    for (int j = 0; j < 8; ++j)                // C/D layout: VGPR j -> row j + 8*half
        out[(size_t)(m0 + j + 8 * half) * 128 + n] = c[j];
}

// ---------------------------------------------------------------- seed agg with self-loop term
// agg = hW * self_norm + bias   (replaces zero-fill; edge atomics accumulate on top)
__global__ __launch_bounds__(256) void agg_init_kernel(
    const float* __restrict__ hW, const float* __restrict__ dis,
    const float* __restrict__ bias, float* __restrict__ agg, int nN) {
    int nidx = blockIdx.x * 8 + (threadIdx.x >> 5);
    if (nidx >= nN) return;
    int lane = threadIdx.x & 31;
    float di = dis[nidx];
    float sn = di * di;
    float4 w  = ((const float4*)(hW + (size_t)nidx * 128))[lane];
    float4 bb = ((const float4*)bias)[lane];
    float4 r;
    r.x = w.x * sn + bb.x;
    r.y = w.y * sn + bb.y;
    r.z = w.z * sn + bb.z;
    r.w = w.w * sn + bb.w;
    ((float4*)(agg + (size_t)nidx * 128))[lane] = r;
}

// ---------------------------------------------------------------- edge scatter-add
// One wave per edge; each lane handles 4 contiguous columns (float4 gather).
__global__ __launch_bounds__(256) void edge_agg_kernel(
    const float* __restrict__ hW, const int* __restrict__ src, const int* __restrict__ dst,
    const float* __restrict__ norm, float* __restrict__ agg, int nE) {
    int e = blockIdx.x * 8 + (threadIdx.x >> 5);
    if (e >= nE) return;
    int lane = threadIdx.x & 31;
    int s = src[e], d = dst[e];
    float nv = norm[e];
    float4 v = ((const float4*)(hW + (size_t)s * 128))[lane];
    float* ap = agg + (size_t)d * 128 + lane * 4;
    atomicAdd(ap + 0, v.x * nv);
    atomicAdd(ap + 1, v.y * nv);
    atomicAdd(ap + 2, v.z * nv);
    atomicAdd(ap + 3, v.w * nv);
}

// ---------------------------------------------------------------- mean-pool accumulation
__global__ __launch_bounds__(256) void pool_kernel(
    const float* __restrict__ h, const int* __restrict__ batch,
    float* __restrict__ sums, float* __restrict__ counts, int nN) {
    int nidx = blockIdx.x * 8 + (threadIdx.x >> 5);
    if (nidx >= nN) return;
    int lane = threadIdx.x & 31;
    int g = batch[nidx];
    float4 v = ((const float4*)(h + (size_t)nidx * 128))[lane];
    float* sp = sums + (size_t)g * 128 + lane * 4;
    atomicAdd(sp + 0, v.x);
    atomicAdd(sp + 1, v.y);
    atomicAdd(sp + 2, v.z);
    atomicAdd(sp + 3, v.w);
    if (lane == 0) atomicAdd(&counts[g], 1.0f);
}

// ---------------------------------------------------------------- head: out[b][j] = [pooled, cond] @ lin_W + lin_b
__global__ void head_kernel(
    const float* __restrict__ sums, const float* __restrict__ counts,
    const float* __restrict__ cond, const float* __restrict__ lin_W,
    const float* __restrict__ lin_b, float* __restrict__ out) {
    int t = threadIdx.x;                       // 128 threads: (graph, class)
    if (t >= N_GRAPHS * N_C) return;
    int g = t >> 1, j = t & 1;
    float inv = 1.0f / fmaxf(counts[g], 1.0f);
    float acc = lin_b[j];
    #pragma unroll 4
    for (int k = 0; k < D_H; ++k)
        acc += (sums[g * D_H + k] * inv) * lin_W[k * N_C + j];
    #pragma unroll
    for (int k = 0; k < D_COND; ++k)
        acc += cond[g * D_COND + k] * lin_W[(D_H + k) * N_C + j];
    out[g * N_C + j] = acc;
}

// ---------------------------------------------------------------- launch
extern "C" void kernel_launch(void* const* d_in, const int* in_sizes, int n_in,
                              void* d_out, int out_size, void* d_ws, size_t ws_size,
                              hipStream_t stream) {
    const float* x     = (const float*)d_in[0];
    const float* cond  = (const float*)d_in[1];
    const float* Ws[4] = {(const float*)d_in[2], (const float*)d_in[4],
                          (const float*)d_in[6], (const float*)d_in[8]};
    const float* bs[4] = {(const float*)d_in[3], (const float*)d_in[5],
                          (const float*)d_in[7], (const float*)d_in[9]};
    const float* lin_W = (const float*)d_in[10];
    const float* lin_b = (const float*)d_in[11];
    const int*   ei    = (const int*)d_in[12];
    const int*   src   = ei;
    const int*   dst   = ei + N_EDGES;
    const int*   batch = (const int*)d_in[13];
    float* out = (float*)d_out;

    // workspace layout (256B aligned)
    char* ws = (char*)d_ws;
    size_t off = 0;
    auto carve = [&](size_t bytes) {
        void* p = ws + off;
        off += (bytes + 255) & ~size_t(255);
        return p;
    };
    float* hW     = (float*)carve((size_t)N_NODES * D_H * 4);   // post-GEMM features
    float* agg    = (float*)carve((size_t)N_NODES * D_H * 4);   // layer output (pre-activation)
    float* dis    = (float*)carve((size_t)N_NODES * 4);
    float* norm   = (float*)carve((size_t)N_EDGES * 4);
    float* sums   = (float*)carve((size_t)N_GRAPHS * D_H * 4);
    float* counts = (float*)carve((size_t)N_GRAPHS * 4);

    const int TB = 256;
    const int edgeBlocks = (N_EDGES + 7) / 8;    // wave per edge, 8 waves/block
    const int nodeBlocks = (N_NODES + 7) / 8;

    // --- normalization pre-pass
    fill_f32<<<(N_NODES + TB - 1) / TB, TB, 0, stream>>>(dis, 1.0f, N_NODES);   // deg = 1
    deg_kernel<<<(N_EDGES + TB - 1) / TB, TB, 0, stream>>>(dst, dis, N_EDGES);
    dis_kernel<<<(N_NODES + TB - 1) / TB, TB, 0, stream>>>(dis, N_NODES);
    norm_kernel<<<(N_EDGES + TB - 1) / TB, TB, 0, stream>>>(src, dst, dis, norm, N_EDGES);

    // --- 4 GCN layers: GEMM (ReLU of prev layer fused into A-load) -> seed agg -> edge atomics
    for (int L = 0; L < 4; ++L) {
        const float* hin = (L == 0) ? x : agg;
        gemm_wmma_f32<<<N_NODES / 16, 256, 0, stream>>>(hin, Ws[L], hW, (L > 0) ? 1 : 0);
        agg_init_kernel<<<nodeBlocks, 256, 0, stream>>>(hW, dis, bs[L], agg, N_NODES);
        edge_agg_kernel<<<edgeBlocks, 256, 0, stream>>>(hW, src, dst, norm, agg, N_EDGES);
    }
    // agg now holds layer-4 output (no activation), exactly as the reference.

    // --- global mean pool + linear head
    fill_f32<<<(N_GRAPHS * D_H + TB - 1) / TB, TB, 0, stream>>>(sums, 0.0f, N_GRAPHS * D_H);
    fill_f32<<<1, TB, 0, stream>>>(counts, 0.0f, N_GRAPHS);
    pool_kernel<<<nodeBlocks, 256, 0, stream>>>(agg, batch, sums, counts, N_NODES);
    head_kernel<<<1, 128, 0, stream>>>(sums, counts, cond, lin_W, lin_b, out);
}